// CompiledEvalLoss_43087111914287
// MI455X (gfx1250) — compile-verified
//
#include <hip/hip_runtime.h>
#include <hip/hip_bf16.h>
#include <math.h>

typedef float v2f __attribute__((ext_vector_type(2)));
typedef float v4f __attribute__((ext_vector_type(4)));
typedef float v8f __attribute__((ext_vector_type(8)));

#define Bn 512
#define NAT 1024
#define Gn 256
#define Ln 8
#define Kn 128
#define Dn 6
#define KT (Kn / 16)

#define KBT        0.59616123f          /* KB * TEMP */
#define ALPHA_C    10.0f
#define INV2BW2    22.222222222222221f  /* 1/(2*0.15^2) */
#define LOG_NORM  (-1.0172542224226055f)/* log(128) + 3*log(2*pi*0.15^2) */

// Row layout: 8 floats, permuted [d0,d1,d4,d5, d2,d3,0,0].
// Lane half=0 reads floats 0..3 -> frag0={d0,d1} (K=0,1), frag1={d4,d5}
// Lane half=1 reads floats 4..7 -> frag0={d2,d3} (K=2,3), frag1={0,0}
// => one aligned ds_load_b128 per operand, zero divergent selects.
__device__ __forceinline__ int slot_of(int d) {
    return d < 2 ? d : (d < 4 ? d + 2 : d - 2);
}

// ---------------------------------------------------------------------------
// Kernel 1: one block per (g,l). Stage permuted dists + kde in LDS, run the
// 512x128x6 einsum on V_WMMA_F32_16X16X4_F32 (D=6 as K=4 + padded K=2),
// then a two-pass logsumexp over the full 8x(8 k-tiles) register tile.
// ---------------------------------------------------------------------------
__global__ __launch_bounds__(128)
void eval_scaled_kernel(const float* __restrict__ pos,
                        const float* __restrict__ kde,
                        const float* __restrict__ weight,
                        const float* __restrict__ offset,
                        const int*   __restrict__ aidx,
                        float* __restrict__ scaled)
{
    __shared__ float s_dist[Bn * 8];    // 16 KB, padded/permuted rows
    __shared__ float s_x2[Bn];          //  2 KB
    __shared__ float s_kde[Kn * 8];     //  4 KB, padded/permuted rows
    __shared__ float s_mu2[Kn];         // 0.5 KB

    const int gl  = blockIdx.x;         // 0 .. G*L-1
    const int g   = gl / Ln;
    const int l   = gl % Ln;
    const int tid = threadIdx.x;        // 0..127

    const int a0 = aidx[gl * 4 + 0];
    const int a1 = aidx[gl * 4 + 1];
    const int a2 = aidx[gl * 4 + 2];
    const int a3 = aidx[gl * 4 + 3];

    // stage kde[g,l] (128x6) permuted into 128x8 rows, slots 6,7 zeroed
    const float* kptr = kde + (size_t)gl * Kn * Dn;
    for (int k = tid; k < Kn; k += 128) {
        float m2 = 0.f;
#pragma unroll
        for (int d = 0; d < Dn; ++d) {
            float v = kptr[k * Dn + d];
            s_kde[k * 8 + slot_of(d)] = v;
            m2 += v * v;
        }
        s_kde[k * 8 + 6] = 0.f;
        s_kde[k * 8 + 7] = 0.f;
        s_mu2[k] = m2;
    }

    // dists[b] -> permuted row of 8, plus x2[b]
    {
        const int PI[6] = {0, 0, 0, 1, 1, 2};
        const int PJ[6] = {1, 2, 3, 2, 3, 3};
        for (int b = tid; b < Bn; b += 128) {
            float px[4], py[4], pz[4];
            const int aa[4] = {a0, a1, a2, a3};
#pragma unroll
            for (int a = 0; a < 4; ++a) {
                const float* pp = pos + ((size_t)b * NAT + aa[a]) * 3;
                px[a] = pp[0]; py[a] = pp[1]; pz[a] = pp[2];
            }
            float x2 = 0.f;
#pragma unroll
            for (int p = 0; p < 6; ++p) {
                float dx = px[PI[p]] - px[PJ[p]];
                float dy = py[PI[p]] - py[PJ[p]];
                float dz = pz[PI[p]] - pz[PJ[p]];
                float d  = sqrtf(dx * dx + dy * dy + dz * dz + 1e-12f);
                s_dist[b * 8 + slot_of(p)] = d;
                x2 += d * d;
            }
            s_dist[b * 8 + 6] = 0.f;
            s_dist[b * 8 + 7] = 0.f;
            s_x2[b] = x2;
        }
    }
    __syncthreads();

    const float w   = weight[gl];
    const float off = offset[gl];

    const int wave = tid >> 5;
    const int lane = tid & 31;
    const int m16  = lane & 15;          // M (A) / N (B,C) index within 16
    const int half = lane >> 4;          // K-half selector for A/B frags

    // Hoist all B fragments + mu2: depend only on k-tile. 8 x b128 loads.
    v4f  Bf[KT];
    float mu2v[KT];
#pragma unroll
    for (int kt = 0; kt < KT; ++kt) {
        const int n = kt * 16 + m16;
        Bf[kt]   = *(const v4f*)&s_kde[n * 8 + half * 4];
        mu2v[kt] = s_mu2[n];
    }

    // each wave processes b-tiles: 32 tiles of 16 rows, 4 waves
    for (int bt = wave; bt < Bn / 16; bt += 4) {
        const v4f Af = *(const v4f*)&s_dist[(bt * 16 + m16) * 8 + half * 4];
        const v2f A0 = {Af.x, Af.y};
        const v2f A1 = {Af.z, Af.w};

        float x2m[8];
#pragma unroll
        for (int r = 0; r < 8; ++r) x2m[r] = s_x2[bt * 16 + r + 8 * half];

        // Pass A: all 64 exponent arguments t[kt][r] into registers
        float t[KT][8];
#pragma unroll
        for (int kt = 0; kt < KT; ++kt) {
            const v2f B0 = {Bf[kt].x, Bf[kt].y};
            const v2f B1 = {Bf[kt].z, Bf[kt].w};

            v8f c = {};
            c = __builtin_amdgcn_wmma_f32_16x16x4_f32(false, A0, false, B0,
                                                      (short)0, c, false, false);
            c = __builtin_amdgcn_wmma_f32_16x16x4_f32(false, A1, false, B1,
                                                      (short)0, c, false, false);

            const float mu2n = mu2v[kt];
#pragma unroll
            for (int r = 0; r < 8; ++r)
                t[kt][r] = (2.f * c[r] - (x2m[r] + mu2n)) * INV2BW2;
        }

        // Pass B: row max over kt, then over the 16 N-lanes (xor stays in halves)
        float rmax[8];
#pragma unroll
        for (int r = 0; r < 8; ++r) {
            float m = t[0][r];
#pragma unroll
            for (int kt = 1; kt < KT; ++kt) m = fmaxf(m, t[kt][r]);
            rmax[r] = m;
        }
#pragma unroll
        for (int d = 1; d < 16; d <<= 1)
#pragma unroll
            for (int r = 0; r < 8; ++r)
                rmax[r] = fmaxf(rmax[r], __shfl_xor(rmax[r], d, 32));

        // Pass C: single exp sweep + sum, then cross-lane sum
        float rsum[8];
#pragma unroll
        for (int r = 0; r < 8; ++r) {
            float s = 0.f;
#pragma unroll
            for (int kt = 0; kt < KT; ++kt) s += __expf(t[kt][r] - rmax[r]);
            rsum[r] = s;
        }
#pragma unroll
        for (int d = 1; d < 16; d <<= 1)
#pragma unroll
            for (int r = 0; r < 8; ++r)
                rsum[r] += __shfl_xor(rsum[r], d, 32);

        if (m16 == 0) {
#pragma unroll
            for (int r = 0; r < 8; ++r) {
                int   b      = bt * 16 + r + 8 * half;
                float logP   = rmax[r] + __logf(rsum[r]) - LOG_NORM;
                float energy = -KBT * logP;
                scaled[((size_t)b * Gn + g) * Ln + l] = energy * w + off;
            }
        }
    }
}

// ---------------------------------------------------------------------------
// Kernel 2: one block (256 threads = G) per batch element.
// masked min over L, then block logsumexp over G.
// ---------------------------------------------------------------------------
__global__ __launch_bounds__(256)
void reduce_out_kernel(const float* __restrict__ scaled,
                       const int*   __restrict__ gsize,
                       float* __restrict__ out)
{
    __shared__ float red[Gn];
    const int b = blockIdx.x;
    const int g = threadIdx.x;

    int sz = gsize[g]; if (sz < 1) sz = 1;
    const float* p = scaled + ((size_t)b * Gn + g) * Ln;
    float mn = INFINITY;
#pragma unroll
    for (int l = 0; l < Ln; ++l)
        if (l < sz) mn = fminf(mn, p[l]);

    float t = -ALPHA_C * mn;

    red[g] = t;
    __syncthreads();
    for (int s = Gn / 2; s > 0; s >>= 1) {
        if (g < s) red[g] = fmaxf(red[g], red[g + s]);
        __syncthreads();
    }
    float mx = red[0];
    __syncthreads();
    red[g] = __expf(t - mx);
    __syncthreads();
    for (int s = Gn / 2; s > 0; s >>= 1) {
        if (g < s) red[g] += red[g + s];
        __syncthreads();
    }
    if (g == 0) out[b] = -(mx + __logf(red[0])) / ALPHA_C;
}

extern "C" void kernel_launch(void* const* d_in, const int* in_sizes, int n_in,
                              void* d_out, int out_size, void* d_ws, size_t ws_size,
                              hipStream_t stream)
{
    const float* pos    = (const float*)d_in[0];  // (B, NAT, 3)
    const float* kde    = (const float*)d_in[1];  // (G, L, K, D)
    const float* weight = (const float*)d_in[2];  // (G, L)
    const float* offset = (const float*)d_in[3];  // (G, L)
    const int*   aidx   = (const int*)d_in[4];    // (G, L, 4)
    const int*   gsize  = (const int*)d_in[5];    // (G,)
    float*       out    = (float*)d_out;          // (B,)
    float*       scaled = (float*)d_ws;           // B*G*L floats = 4 MiB

    eval_scaled_kernel<<<Gn * Ln, 128, 0, stream>>>(pos, kde, weight, offset,
                                                    aidx, scaled);
    reduce_out_kernel<<<Bn, Gn, 0, stream>>>(scaled, gsize, out);
}